// FFM_Layer_7902739825046
// MI455X (gfx1250) — compile-verified
//
#include <hip/hip_runtime.h>

typedef __attribute__((ext_vector_type(2))) float v2f;
typedef __attribute__((ext_vector_type(8))) float v8f;

#define N_DENSE  13
#define N_SPARSE 26
#define ONEHOT   1000
#define FIELD    39            // N_DENSE + N_SPARSE
#define ROW      312           // FIELD*8 contiguous floats per v row
#define ROWQ     78            // float4 per v row (312*4B = 78*16B, aligned)
#define PAD      340           // ldsD row pitch (dwords): 16B-aligned, bank-safe
#define BPITCH   336           // ldsB2 pair pitch: 2*336 = 672 dwords = 32-bank shift
#define BATCH    4096

// One workgroup = 128 threads = 4 wave32 = 16 samples.
__global__ __launch_bounds__(128) void ffm_fused_kernel(
    const float* __restrict__ inputs,   // (B, 39)
    const float* __restrict__ w0,       // (1,)
    const float* __restrict__ w,        // (26013,)
    const float* __restrict__ v,        // (26013, 39, 8)
    float* __restrict__ out)            // (B, 1)
{
    // Row-pair interleaved: element (r,c) at ((r>>1)*PITCH + c)*2 + (r&1)
    // -> WMMA operand (r0, r0+1) is one contiguous 8B ds_load_b64.
    __shared__ float ldsB2[8 * BPITCH * 2];   // zero-padded v[:13] block
    __shared__ float ldsA2[8 * 16 * 2];       // dense 16x16 block
    __shared__ float ldsD[16 * PAD];          // dense field_f for 16 samples

    const int tid  = threadIdx.x;
    const int lane = tid & 31;
    const int wave = tid >> 5;
    const int b0   = blockIdx.x * 16;

    // -------- Stage 0: branch-free LDS staging (clamped loads + selects) ----
    for (int e = tid; e < 8 * BPITCH * 2; e += 128) {
        const int rp   = e / (2 * BPITCH);
        const int rem  = e - rp * (2 * BPITCH);
        const int c    = rem >> 1;
        const int r    = 2 * rp + (rem & 1);
        const bool ok  = (r < N_DENSE) & (c < ROW);
        const float val = v[ok ? (r * ROW + c) : 0];   // unconditional load
        ldsB2[e] = ok ? val : 0.0f;                    // v_cndmask, no branch
    }
    for (int e = tid; e < 8 * 16 * 2; e += 128) {
        const int kp = e >> 5;                 // pair index
        const int mm = (e >> 1) & 15;
        const int k  = 2 * kp + (e & 1);
        const float val = inputs[(b0 + mm) * FIELD + ((k < N_DENSE) ? k : 0)];
        ldsA2[e] = (k < N_DENSE) ? val : 0.0f;
    }
    __syncthreads();

    // -------- Stage 1: dense GEMM via V_WMMA_F32_16X16X4_F32 ---------------
    // A 16x4 step layout: lanes 0-15 hold K=4s,4s+1; lanes 16-31 K=4s+2,4s+3.
    const int m     = lane & 15;
    const int ph    = lane >> 4;               // pair-half select (0 or 1)
    v2f a[4];
#pragma unroll
    for (int s = 0; s < 4; ++s)
        a[s] = *(const v2f*)&ldsA2[((2 * s + ph) * 16 + m) * 2];

#pragma unroll
    for (int tt = 0; tt < 5; ++tt) {
        const int c = (wave * 5 + tt) * 16 + (lane & 15);
        v8f acc = {};
#pragma unroll
        for (int s = 0; s < 4; ++s) {
            const v2f bb = *(const v2f*)&ldsB2[((2 * s + ph) * BPITCH + c) * 2];
            acc = __builtin_amdgcn_wmma_f32_16x16x4_f32(
                false, a[s], false, bb, (short)0, acc, false, false);
        }
        // D layout: VGPR r: lanes 0-15 -> M=r, lanes 16-31 -> M=8+r.
        const int mrow = ph * 8;
#pragma unroll
        for (int r = 0; r < 8; ++r)
            ldsD[(mrow + r) * PAD + c] = acc[r];
    }
    __syncthreads();

    // -------- Stage 2: float4 sparse gather + wave reductions --------------
    // Lane owns float4 slots p = lane, lane+32 (+ masked tail slot).
    // 4p mod 8 is constant per lane -> component j maps to fixed k slot.
    const float4* __restrict__ v4 = (const float4*)v;
    const bool  tail_ok = lane < (ROWQ - 64);          // 14 tail slots
    const float tmask   = tail_ok ? 1.0f : 0.0f;
    const int   p2      = tail_ok ? (lane + 64) : (ROWQ - 1);

#pragma unroll 1
    for (int j = 0; j < 4; ++j) {
        const int ms = wave * 4 + j;
        const int b  = b0 + ms;

        float4 f0 = *(const float4*)&ldsD[ms * PAD + 4 * lane];
        float4 f1 = *(const float4*)&ldsD[ms * PAD + 4 * (lane + 32)];
        float4 f2 = *(const float4*)&ldsD[ms * PAD + 4 * p2];
        f2.x *= tmask; f2.y *= tmask; f2.z *= tmask; f2.w *= tmask;

        // preload sparse row bases so gathers can be pipelined
        int rb[N_SPARSE];
#pragma unroll
        for (int f = 0; f < N_SPARSE; ++f) {
            const int idx = (int)inputs[b * FIELD + N_DENSE + f];
            rb[f] = (N_DENSE + f * ONEHOT + idx) * ROWQ;   // float4 units
        }

#pragma unroll 2
        for (int f = 0; f < N_SPARSE; ++f) {
            const float4 r0 = v4[rb[f] + lane];        // global_load_b128
            const float4 r1 = v4[rb[f] + lane + 32];
            const float4 r2 = v4[rb[f] + p2];          // clamped, masked
            f0.x += r0.x; f0.y += r0.y; f0.z += r0.z; f0.w += r0.w;
            f1.x += r1.x; f1.y += r1.y; f1.z += r1.z; f1.w += r1.w;
            f2.x = fmaf(tmask, r2.x, f2.x);
            f2.y = fmaf(tmask, r2.y, f2.y);
            f2.z = fmaf(tmask, r2.z, f2.z);
            f2.w = fmaf(tmask, r2.w, f2.w);
        }

        float4 ps;
        ps.x = f0.x + f1.x + f2.x;
        ps.y = f0.y + f1.y + f2.y;
        ps.z = f0.z + f1.z + f2.z;
        ps.w = f0.w + f1.w + f2.w;

        float psq = f0.x*f0.x + f0.y*f0.y + f0.z*f0.z + f0.w*f0.w
                  + f1.x*f1.x + f1.y*f1.y + f1.z*f1.z + f1.w*f1.w
                  + f2.x*f2.x + f2.y*f2.y + f2.z*f2.z + f2.w*f2.w;
#pragma unroll
        for (int off = 16; off > 0; off >>= 1)
            psq += __shfl_xor(psq, off, 32);           // ||field||^2, all lanes

        // parity-preserving reduce: even lanes -> (s0..s3), odd -> (s4..s7)
#pragma unroll
        for (int off = 2; off <= 16; off <<= 1) {
            ps.x += __shfl_xor(ps.x, off, 32);
            ps.y += __shfl_xor(ps.y, off, 32);
            ps.z += __shfl_xor(ps.z, off, 32);
            ps.w += __shfl_xor(ps.w, off, 32);
        }
        float sq = ps.x*ps.x + ps.y*ps.y + ps.z*ps.z + ps.w*ps.w;
        sq += __shfl_xor(sq, 1, 32);                   // sum_k s_k^2, all lanes

        const float inter = 0.5f * (sq - psq);

        // linear part: clamped unconditional loads + selects (no exec branch)
        const int lc = (lane < N_DENSE) ? lane : 0;
        float lin = inputs[b * FIELD + lc] * w[lc];
        lin = (lane < N_DENSE) ? lin : 0.0f;
        const int ls   = (lane < N_SPARSE) ? lane : 0;
        const int sidx = (int)inputs[b * FIELD + N_DENSE + ls];
        const float sterm = w[N_DENSE + ls * ONEHOT + sidx];
        lin += (lane < N_SPARSE) ? sterm : 0.0f;
#pragma unroll
        for (int off = 16; off > 0; off >>= 1)
            lin += __shfl_xor(lin, off, 32);

        if (lane == 0)
            out[b] = w0[0] + lin + inter;
    }
}

extern "C" void kernel_launch(void* const* d_in, const int* in_sizes, int n_in,
                              void* d_out, int out_size, void* d_ws, size_t ws_size,
                              hipStream_t stream) {
    const float* inputs = (const float*)d_in[0];   // (4096, 39)
    const float* w0     = (const float*)d_in[1];   // (1,)
    const float* w      = (const float*)d_in[2];   // (26013,)
    const float* v      = (const float*)d_in[3];   // (26013, 39, 8)
    float* out          = (float*)d_out;           // (4096, 1)

    (void)in_sizes; (void)n_in; (void)out_size; (void)d_ws; (void)ws_size;

    dim3 grid(BATCH / 16);   // 256 workgroups, 16 samples each
    dim3 block(128);         // 4 wave32
    ffm_fused_kernel<<<grid, block, 0, stream>>>(inputs, w0, w, v, out);
}